// InductiveBundleMapLearner_51049981280276
// MI455X (gfx1250) — compile-verified
//
#include <hip/hip_runtime.h>
#include <hip/hip_bf16.h>

typedef float v2f __attribute__((ext_vector_type(2)));
typedef float v8f __attribute__((ext_vector_type(8)));

#define NN 50000
#define EE 800000
#define IN_DIM 512
#define H1 256
#define H2 128
#define NP 28

// ---------------------------------------------------------------------------
// Degree: one thread per edge, atomic add 1.0 to deg[dst]
// ---------------------------------------------------------------------------
__global__ __launch_bounds__(256) void deg_kernel(const long long* __restrict__ ei,
                                                  float* __restrict__ deg, int E) {
    int e = blockIdx.x * blockDim.x + threadIdx.x;
    if (e >= E) return;
    int dst = (int)ei[(size_t)E + e];
    unsafeAtomicAdd(deg + dst, 1.0f);
}

// deg[i] <- 1 / max(deg[i], 1)  (in place)
__global__ __launch_bounds__(256) void invdeg_kernel(float* __restrict__ deg, int n) {
    int i = blockIdx.x * blockDim.x + threadIdx.x;
    if (i >= n) return;
    deg[i] = 1.0f / fmaxf(deg[i], 1.0f);
}

// ---------------------------------------------------------------------------
// Y[M, Nout] = X[M, K] @ W[Nout, K]^T  (+ bias), fp32 WMMA 16x16x4.
// One wave32 per 16x16 output tile. M, K, Nout all multiples of 16.
//
// A-frag layout (16x4 f32): lanes 0-15 hold M=lane, VGPR0/1 = K0/K1;
//                           lanes16-31 hold M=lane-16, VGPR0/1 = K2/K3.
// B-frag layout (4x16 f32): VGPR0 lanes0-15 = row K0 (N=lane), lanes16-31 = K2;
//                           VGPR1 lanes0-15 = K1, lanes16-31 = K3.
// With B = W^T both frags read float2 at (row*ld + k + 2*half).
// C/D: VGPR r -> row r (lanes 0-15) / row r+8 (lanes 16-31), N = lane&15.
// ---------------------------------------------------------------------------
__global__ __launch_bounds__(256) void gemm_wmma_f32(const float* __restrict__ X,
                                                     const float* __restrict__ W,
                                                     const float* __restrict__ bias,
                                                     float* __restrict__ Y,
                                                     int M, int K, int Nout,
                                                     int total_tiles) {
    int tile = blockIdx.x * 8 + (threadIdx.x >> 5);   // 8 waves per block
    if (tile >= total_tiles) return;                  // wave-uniform: EXEC stays full
    int lane = threadIdx.x & 31;
    int tiles_n = Nout >> 4;
    int tm = tile / tiles_n;
    int tn = tile - tm * tiles_n;

    int half = lane >> 4;       // 0: lanes 0-15, 1: lanes 16-31
    int r    = lane & 15;
    int koff = half * 2;

    const float* xrow = X + (size_t)(tm * 16 + r) * K;
    const float* wrow = W + (size_t)(tn * 16 + r) * K;

    v8f c = {};
    for (int k = 0; k < K; k += 16) {
#pragma unroll
        for (int s = 0; s < 4; ++s) {
            v2f a = *(const v2f*)(xrow + k + s * 4 + koff);
            v2f b = *(const v2f*)(wrow + k + s * 4 + koff);
            c = __builtin_amdgcn_wmma_f32_16x16x4_f32(
                    /*neg_a=*/false, a, /*neg_b=*/false, b,
                    /*c_mod=*/(short)0, c, /*reuse_a=*/false, /*reuse_b=*/false);
        }
    }

    float bv = (bias != nullptr) ? bias[tn * 16 + r] : 0.0f;
    float* yp = Y + (size_t)(tm * 16 + half * 8) * Nout + tn * 16 + r;
#pragma unroll
    for (int i = 0; i < 8; ++i)
        yp[(size_t)i * Nout] = c[i] + bv;
}

// ---------------------------------------------------------------------------
// Scatter-add: G[dst, :] += T[src, :] over all edges. F multiple of 4.
// Consecutive threads = consecutive float4s of the same edge (coalesced).
// ---------------------------------------------------------------------------
__global__ __launch_bounds__(256) void scatter_kernel(const long long* __restrict__ ei,
                                                      const float* __restrict__ T,
                                                      float* __restrict__ G,
                                                      int E, int F) {
    int groups = F >> 2;
    long long tid = (long long)blockIdx.x * blockDim.x + threadIdx.x;
    if (tid >= (long long)E * groups) return;
    int e = (int)(tid / groups);
    int g = (int)(tid - (long long)e * groups);
    int src = (int)ei[e];
    int dst = (int)ei[(size_t)E + e];
    const float4 v = *(const float4*)(T + (size_t)src * F + g * 4);
    float* out = G + (size_t)dst * F + g * 4;
    unsafeAtomicAdd(out + 0, v.x);
    unsafeAtomicAdd(out + 1, v.y);
    unsafeAtomicAdd(out + 2, v.z);
    unsafeAtomicAdd(out + 3, v.w);
}

// h = relu(G * invdeg[row] + R), written back into G
__global__ __launch_bounds__(256) void combine_kernel(float* __restrict__ G,
                                                      const float* __restrict__ R,
                                                      const float* __restrict__ invdeg,
                                                      int n, int F) {
    long long idx = (long long)blockIdx.x * blockDim.x + threadIdx.x;
    if (idx >= (long long)n * F) return;
    int node = (int)(idx / F);
    float v = G[idx] * invdeg[node] + R[idx];
    G[idx] = fmaxf(v, 0.0f);
}

// params[node, p] = bp[p] + h2[node, :] . Wp[p, :]   (28 outputs, tiny GEMM)
__global__ __launch_bounds__(256) void params_kernel(const float* __restrict__ H,
                                                     const float* __restrict__ Wp,
                                                     const float* __restrict__ bp,
                                                     float* __restrict__ P, int n) {
    int idx = blockIdx.x * blockDim.x + threadIdx.x;
    if (idx >= n * NP) return;
    int node = idx / NP;
    int p = idx - node * NP;
    const float* h = H + (size_t)node * H2;
    const float* w = Wp + (size_t)p * H2;
    float acc = bp[p];
#pragma unroll 4
    for (int k = 0; k < H2; k += 4)
        acc += h[k] * w[k] + h[k + 1] * w[k + 1] + h[k + 2] * w[k + 2] + h[k + 3] * w[k + 3];
    P[idx] = acc;
}

// ---------------------------------------------------------------------------
// Cayley: A skew from 28 params, O = (I-A)^{-1}(I+A) = 2*(I-A)^{-1} - I.
// I-A has symmetric part I (positive definite) -> pivot-free GJ sweeps stable.
// One thread per node, fully unrolled in registers.
// ---------------------------------------------------------------------------
__global__ __launch_bounds__(256) void cayley_kernel(const float* __restrict__ P,
                                                     float* __restrict__ out, int n) {
    int i = blockIdx.x * blockDim.x + threadIdx.x;
    if (i >= n) return;
    const float* p = P + (size_t)i * NP;
    float M[8][8];
    int t = 0;
#pragma unroll
    for (int a = 0; a < 8; ++a) {
        M[a][a] = 1.0f;
#pragma unroll
        for (int b = a + 1; b < 8; ++b) {
            float v = p[t++];
            M[a][b] = -v;   // (I - A): A[a][b] = +v for a<b
            M[b][a] = v;    //          A[b][a] = -v
        }
    }
    // in-place Gauss-Jordan sweeps -> M becomes M^{-1}
#pragma unroll
    for (int k = 0; k < 8; ++k) {
        float pk = 1.0f / M[k][k];
#pragma unroll
        for (int j = 0; j < 8; ++j)
            if (j != k) M[k][j] *= pk;
#pragma unroll
        for (int r = 0; r < 8; ++r) {
            if (r == k) continue;
            float f = M[r][k];
#pragma unroll
            for (int j = 0; j < 8; ++j)
                if (j != k) M[r][j] -= f * M[k][j];
            M[r][k] = -f * pk;
        }
        M[k][k] = pk;
    }
    float* o = out + (size_t)i * 64;
#pragma unroll
    for (int a = 0; a < 8; ++a)
#pragma unroll
        for (int b = 0; b < 8; ++b)
            o[a * 8 + b] = 2.0f * M[a][b] - (a == b ? 1.0f : 0.0f);
}

// ---------------------------------------------------------------------------
extern "C" void kernel_launch(void* const* d_in, const int* in_sizes, int n_in,
                              void* d_out, int out_size, void* d_ws, size_t ws_size,
                              hipStream_t stream) {
    const float*     x   = (const float*)d_in[0];
    const long long* ei  = (const long long*)d_in[1];
    const float*     W1l = (const float*)d_in[2];
    const float*     b1  = (const float*)d_in[3];
    const float*     W1r = (const float*)d_in[4];
    const float*     W2l = (const float*)d_in[5];
    const float*     b2  = (const float*)d_in[6];
    const float*     W2r = (const float*)d_in[7];
    const float*     Wp  = (const float*)d_in[8];
    const float*     bp  = (const float*)d_in[9];
    float* out = (float*)d_out;

    // scratch layout (floats): deg | T1 (N*256) | R1 (N*256) | G1 (N*256) | A2 (N*128)
    float* ws  = (float*)d_ws;
    float* deg = ws;
    float* T1  = deg + NN;                 // t1, later t2
    float* R1  = T1 + (size_t)NN * H1;     // r1, later r2, later params
    float* G1  = R1 + (size_t)NN * H1;     // agg1 -> h1 (in place)
    float* A2  = G1 + (size_t)NN * H1;     // agg2 -> h2 (in place)

    hipMemsetAsync(deg, 0, (size_t)NN * sizeof(float), stream);
    hipMemsetAsync(G1, 0, (size_t)NN * H1 * sizeof(float), stream);
    hipMemsetAsync(A2, 0, (size_t)NN * H2 * sizeof(float), stream);

    // degree -> inverse degree
    deg_kernel<<<(EE + 255) / 256, 256, 0, stream>>>(ei, deg, EE);
    invdeg_kernel<<<(NN + 255) / 256, 256, 0, stream>>>(deg, NN);

    // ---- layer 1: transform-then-aggregate ----
    {
        int tiles = (NN / 16) * (H1 / 16);       // 50000
        int blocks = (tiles + 7) / 8;
        gemm_wmma_f32<<<blocks, 256, 0, stream>>>(x, W1l, nullptr, T1, NN, IN_DIM, H1, tiles);
        gemm_wmma_f32<<<blocks, 256, 0, stream>>>(x, W1r, b1,      R1, NN, IN_DIM, H1, tiles);
    }
    {
        long long tot = (long long)EE * (H1 / 4);
        scatter_kernel<<<(unsigned)((tot + 255) / 256), 256, 0, stream>>>(ei, T1, G1, EE, H1);
        long long ne = (long long)NN * H1;
        combine_kernel<<<(unsigned)((ne + 255) / 256), 256, 0, stream>>>(G1, R1, deg, NN, H1);
    }

    // ---- layer 2 ----
    {
        int tiles = (NN / 16) * (H2 / 16);       // 25000
        int blocks = (tiles + 7) / 8;
        gemm_wmma_f32<<<blocks, 256, 0, stream>>>(G1, W2l, nullptr, T1, NN, H1, H2, tiles);
        gemm_wmma_f32<<<blocks, 256, 0, stream>>>(G1, W2r, b2,      R1, NN, H1, H2, tiles);
    }
    {
        long long tot = (long long)EE * (H2 / 4);
        scatter_kernel<<<(unsigned)((tot + 255) / 256), 256, 0, stream>>>(ei, T1, A2, EE, H2);
        long long ne = (long long)NN * H2;
        combine_kernel<<<(unsigned)((ne + 255) / 256), 256, 0, stream>>>(A2, R1, deg, NN, H2);
    }

    // ---- projection + Cayley ----
    params_kernel<<<(NN * NP + 255) / 256, 256, 0, stream>>>(A2, Wp, bp, R1, NN);
    cayley_kernel<<<(NN + 255) / 256, 256, 0, stream>>>(R1, out, NN);
}